// GLMMD_75256416960850
// MI455X (gfx1250) — compile-verified
//
#include <hip/hip_runtime.h>

typedef __attribute__((ext_vector_type(2))) float v2f;
typedef __attribute__((ext_vector_type(8))) float v8f;
typedef int int4v __attribute__((ext_vector_type(4)));
typedef int4v __attribute__((address_space(1)))* g_int4p;  // global int4*
typedef int4v __attribute__((address_space(3)))* l_int4p;  // LDS int4*

#define B_ROWS 8192
#define D_DIM  2048
#define NCLS   32
#define TM 128
#define TN 128
#define KB 16
#define LDSS 20   // padded LDS row stride (floats): conflict-free AND 16B-aligned rows

// CDNA5 async global->LDS copies (ASYNCcnt) when the toolchain declares them.
#if defined(__AMDGCN__) && __has_builtin(__builtin_amdgcn_global_load_async_to_lds_b128) && \
    __has_builtin(__builtin_amdgcn_s_wait_asynccnt)
#define USE_ASYNC_LDS 1
#else
#define USE_ASYNC_LDS 0
#endif

// ---------------------------------------------------------------------------
// Row norms of gated features: nrm[y][row] = sum_k (f[row][k]*gate[k])^2
// ---------------------------------------------------------------------------
__global__ __launch_bounds__(256) void glmmd_norms(const float* __restrict__ fh,
                                                   const float* __restrict__ fl,
                                                   const float* __restrict__ gate,
                                                   float* __restrict__ nrm) {
  const float* f = blockIdx.y ? fl : fh;
  float* out = nrm + (size_t)blockIdx.y * B_ROWS;
  const int row = blockIdx.x;
  const int t = threadIdx.x;
  const float* rp = f + (size_t)row * D_DIM;
  float s = 0.f;
  for (int k = t; k < D_DIM; k += 256) {
    float v = rp[k] * gate[k];
    s += v * v;
  }
  __shared__ float red[256];
  red[t] = s;
  __syncthreads();
  for (int off = 128; off > 0; off >>= 1) {
    if (t < off) red[t] += red[t + off];
    __syncthreads();
  }
  if (t == 0) out[row] = red[0];
}

// ---------------------------------------------------------------------------
// Bit-pack class masks: bit c of m[row] = (labels[row][c] > 0)
// ---------------------------------------------------------------------------
__global__ __launch_bounds__(256) void glmmd_masks(const int* __restrict__ lh,
                                                   const int* __restrict__ ll,
                                                   unsigned* __restrict__ mh,
                                                   unsigned* __restrict__ ml) {
  const int row = blockIdx.x * 256 + threadIdx.x;
  if (row >= B_ROWS) return;
  const int* ph = lh + (size_t)row * NCLS;
  const int* pl = ll + (size_t)row * NCLS;
  unsigned a = 0u, b = 0u;
#pragma unroll
  for (int c = 0; c < NCLS; c++) {
    a |= (ph[c] > 0) ? (1u << c) : 0u;
    b |= (pl[c] > 0) ? (1u << c) : 0u;
  }
  mh[row] = a;
  ml[row] = b;
}

__global__ void glmmd_zero(float* __restrict__ S) {
  if (threadIdx.x < 3 * NCLS) S[threadIdx.x] = 0.f;
}

// ---------------------------------------------------------------------------
// Fused tile kernel: 128x128 tile of K_{which}; fp32 WMMA GEMM over D=2048,
// double-buffered LDS, async L2->LDS staging (CDNA5) when available, RBF +
// masked per-class bilinear reduction fused in-register.
// Raw features are staged; gate^2 is folded into the A fragments
// (dot = sum_k x*y*g^2 == sum_k (x*g)*(y*g)).
//   which: 0 = (fh,fh)  1 = (fl,fl)  2 = (fh,fl)
// ---------------------------------------------------------------------------
__global__ __launch_bounds__(256) void glmmd_tiles(
    const float* __restrict__ Fh, const float* __restrict__ Fl,
    const float* __restrict__ gate,
    const float* __restrict__ nrmH, const float* __restrict__ nrmL,
    const unsigned* __restrict__ mH, const unsigned* __restrict__ mL,
    float* __restrict__ S) {
  __shared__ float g2_s[D_DIM];
  __shared__ float Ab[2][TM * LDSS];
  __shared__ float Bb[2][TN * LDSS];
  __shared__ float cls_s[NCLS];

  const int which = blockIdx.z;
  const float* X = (which == 1) ? Fl : Fh;
  const float* Y = (which == 0) ? Fh : Fl;
  const float* xn = (which == 1) ? nrmL : nrmH;
  const float* yn = (which == 0) ? nrmH : nrmL;
  const unsigned* mA = (which == 1) ? mL : mH;
  const unsigned* mB = (which == 0) ? mH : mL;

  const int i0 = blockIdx.x * TM;
  const int j0 = blockIdx.y * TN;
  const int t = threadIdx.x;
  const int lane = t & 31;
  const int wave = t >> 5;
  const int wm = wave >> 2;  // 0..1 : which 64-row block
  const int wn = wave & 3;   // 0..3 : which 32-col block
  const int lrow = lane & 15;
  const int lhi = lane >> 4;

  for (int k = t; k < D_DIM; k += 256) {
    const float g = gate[k];
    g2_s[k] = g * g;
  }
  if (t < NCLS) cls_s[t] = 0.f;

  // per-thread staging chunks (two 16B chunks each for A and B):
  // chunk c covers row c>>2, k-offset (c&3)*4; this thread owns c = t, t+256
  const int r0 = t >> 2;
  const int r1 = (t + 256) >> 2;
  const int kk = (t & 3) << 2;
  const float* Xp0 = X + (size_t)(i0 + r0) * D_DIM + kk;
  const float* Xp1 = X + (size_t)(i0 + r1) * D_DIM + kk;
  const float* Yp0 = Y + (size_t)(j0 + r0) * D_DIM + kk;
  const float* Yp1 = Y + (size_t)(j0 + r1) * D_DIM + kk;
  const int lo0 = r0 * LDSS + kk;
  const int lo1 = r1 * LDSS + kk;

  v8f acc[4][2] = {};

#if USE_ASYNC_LDS
  // CDNA5: direct L2 -> LDS async copies, no VGPR round-trip.
#define ASYNC_CP(ldsP, gP)                                                    \
  __builtin_amdgcn_global_load_async_to_lds_b128(                             \
      (g_int4p)(void*)(gP), (l_int4p)(void*)(ldsP), 0, 0)
  auto prefetch = [&](int buf, int k0) {
    ASYNC_CP(&Ab[buf][lo0], Xp0 + k0);
    ASYNC_CP(&Ab[buf][lo1], Xp1 + k0);
    ASYNC_CP(&Bb[buf][lo0], Yp0 + k0);
    ASYNC_CP(&Bb[buf][lo1], Yp1 + k0);
  };
  prefetch(0, 0);
#else
  float4 ra0, ra1, rb0, rb1;
  auto gload = [&](int k0) {
    ra0 = *(const float4*)(Xp0 + k0);
    ra1 = *(const float4*)(Xp1 + k0);
    rb0 = *(const float4*)(Yp0 + k0);
    rb1 = *(const float4*)(Yp1 + k0);
  };
  gload(0);
#endif

  for (int k0 = 0; k0 < D_DIM; k0 += KB) {
    const int cur = (k0 >> 4) & 1;
#if USE_ASYNC_LDS
    __builtin_amdgcn_s_wait_asynccnt(0);  // own copies into buf[cur] landed
    __syncthreads();                      // everyone's copies visible
    if (k0 + KB < D_DIM) prefetch(cur ^ 1, k0 + KB);
#else
    *(float4*)&Ab[cur][lo0] = ra0;
    *(float4*)&Ab[cur][lo1] = ra1;
    *(float4*)&Bb[cur][lo0] = rb0;
    *(float4*)&Bb[cur][lo1] = rb1;
    __syncthreads();
    if (k0 + KB < D_DIM) gload(k0 + KB);  // overlap next fetch with compute
#endif
    // fp32 WMMA: 4 K-steps of 4, 8 accumulators per wave; gate^2 on A frags
#pragma unroll
    for (int ks = 0; ks < KB; ks += 4) {
      const v2f g2 = *(const v2f*)&g2_s[k0 + ks + 2 * lhi];
      v2f af[4], bf[2];
#pragma unroll
      for (int mi = 0; mi < 4; mi++) {
        const int r = wm * 64 + mi * 16 + lrow;
        af[mi] = *(const v2f*)&Ab[cur][r * LDSS + ks + 2 * lhi] * g2;
      }
#pragma unroll
      for (int ni = 0; ni < 2; ni++) {
        const int r = wn * 32 + ni * 16 + lrow;
        bf[ni] = *(const v2f*)&Bb[cur][r * LDSS + ks + 2 * lhi];
      }
#pragma unroll
      for (int mi = 0; mi < 4; mi++)
#pragma unroll
        for (int ni = 0; ni < 2; ni++)
          acc[mi][ni] = __builtin_amdgcn_wmma_f32_16x16x4_f32(
              false, af[mi], false, bf[ni], (short)0, acc[mi][ni], false, false);
    }
  }

  // ---- fused RBF + masked per-class bilinear reduction ----
  // C/D layout: VGPR v, lane L  ->  M = v + 8*(L>>4), N = L&15
  float scls[NCLS];
#pragma unroll
  for (int c = 0; c < NCLS; c++) scls[c] = 0.f;

  const int mbase = i0 + wm * 64;
  const int nbase = j0 + wn * 32;
  float ynv[2];
  unsigned mbv[2];
#pragma unroll
  for (int ni = 0; ni < 2; ni++) {
    const int gj = nbase + ni * 16 + lrow;
    ynv[ni] = yn[gj];
    mbv[ni] = mB[gj];
  }
#pragma unroll
  for (int mi = 0; mi < 4; mi++) {
#pragma unroll
    for (int v = 0; v < 8; v++) {
      const int gi = mbase + mi * 16 + v + 8 * lhi;
      const float xnv = xn[gi];
      const unsigned mav = mA[gi];
#pragma unroll
      for (int ni = 0; ni < 2; ni++) {
        const float dist = xnv + ynv[ni] - 2.f * acc[mi][ni][v];
        const float kv = __expf(-0.5f * dist);  // sigma = 1.0
        const unsigned m = mav & mbv[ni];
#pragma unroll
        for (int c = 0; c < NCLS; c++)
          scls[c] += ((m >> c) & 1u) ? kv : 0.f;
      }
    }
  }

  // butterfly reduce across the wave; lane L ends up holding class-L total
  float mytot = 0.f;
#pragma unroll
  for (int cc = 0; cc < NCLS; cc++) {
    float v = scls[cc];
    v += __shfl_xor(v, 16);
    v += __shfl_xor(v, 8);
    v += __shfl_xor(v, 4);
    v += __shfl_xor(v, 2);
    v += __shfl_xor(v, 1);
    if (lane == cc) mytot = v;
  }
  atomicAdd(&cls_s[lane], mytot);  // 32 distinct addresses per wave: conflict-free
  __syncthreads();
  if (t < NCLS) atomicAdd(&S[which * NCLS + t], cls_s[t]);
}

// ---------------------------------------------------------------------------
// Finalize: per-class counts via mask popcount scan + weighted combination
// ---------------------------------------------------------------------------
__global__ void glmmd_final(const unsigned* __restrict__ mh,
                            const unsigned* __restrict__ ml,
                            const float* __restrict__ S,
                            float* __restrict__ out) {
  const int c = threadIdx.x;  // 32 threads, one per class
  float nh = 0.f, nl = 0.f;
  for (int i = 0; i < B_ROWS; i++) {
    nh += (float)((mh[i] >> c) & 1u);
    nl += (float)((ml[i] >> c) & 1u);
  }
  const bool valid = (nh > 0.f) && (nl > 0.f);
  const float nhs = valid ? nh : 1.f;
  const float nls = valid ? nl : 1.f;
  float l = S[c] / (nhs * nhs) + S[NCLS + c] / (nls * nls)
          - 2.f * S[2 * NCLS + c] / (nhs * nls);
  l = valid ? l : 0.f;
#pragma unroll
  for (int off = 16; off > 0; off >>= 1) l += __shfl_down(l, off);
  if (c == 0) out[0] = l / (float)NCLS;
}

// ---------------------------------------------------------------------------
extern "C" void kernel_launch(void* const* d_in, const int* in_sizes, int n_in,
                              void* d_out, int out_size, void* d_ws, size_t ws_size,
                              hipStream_t stream) {
  const float* fh = (const float*)d_in[0];
  const float* fl = (const float*)d_in[1];
  const int* lh = (const int*)d_in[2];
  const int* ll = (const int*)d_in[3];
  const float* gate = (const float*)d_in[4];
  float* out = (float*)d_out;

  // workspace layout (~131 KB)
  float* nrm = (float*)d_ws;                       // [2][B_ROWS]
  unsigned* mh = (unsigned*)(nrm + 2 * B_ROWS);    // [B_ROWS]
  unsigned* ml = mh + B_ROWS;                      // [B_ROWS]
  float* S = (float*)(ml + B_ROWS);                // [3][NCLS]

  glmmd_zero<<<dim3(1), dim3(128), 0, stream>>>(S);
  glmmd_norms<<<dim3(B_ROWS, 2), dim3(256), 0, stream>>>(fh, fl, gate, nrm);
  glmmd_masks<<<dim3(B_ROWS / 256), dim3(256), 0, stream>>>(lh, ll, mh, ml);
  glmmd_tiles<<<dim3(B_ROWS / TM, B_ROWS / TN, 3), dim3(256), 0, stream>>>(
      fh, fl, gate, nrm, nrm + B_ROWS, mh, ml, S);
  glmmd_final<<<dim3(1), dim3(32), 0, stream>>>(mh, ml, S, out);
}